// LiftSplatShoot_13657996001570
// MI455X (gfx1250) — compile-verified
//
#include <hip/hip_runtime.h>
#include <hip/hip_bf16.h>

// ---------------------------------------------------------------------------
// Lift-Splat-Shoot forward for MI455X (gfx1250), wave32 + V_WMMA_F32_16X16X4_F32
// Branch-free K-loops (padded activations + packed weights), dual-N-tile waves
// for A-fragment reuse, branch-free depthnet epilogue via padded y/bias.
// ---------------------------------------------------------------------------

typedef __attribute__((ext_vector_type(2))) float v2f;
typedef __attribute__((ext_vector_type(8))) float v8f;

#define NCAM 6
#define INC_ 256
#define H_ 56
#define W_ 100
#define HW_ 5600          // H_*W_
#define DD_ 41
#define CC_ 64
#define OC_ 105           // DD_+CC_
#define OCP_ 112          // OC_ padded to 7*16
#define NXY_ 33
#define NPIX_ 1089        // 33*33
#define NZ_ 2
#define CZ_ 128           // CC_*NZ_
#define PW_ 35            // padded width  (33 + halo)
#define PPIX_ 1225        // 35*35
#define POFF_ 36          // 1*35 + 1: offset of (0,0) inside padded map

// ---------------------------------------------------------------------------
// Prologue: pad depthnet weights to 112x256 and bias to 112 (zero rows >=105)
// ---------------------------------------------------------------------------
__global__ void lss_pack_dnw(const float* __restrict__ dn_w, const float* __restrict__ dn_b,
                             float* __restrict__ wpad, float* __restrict__ bpad)
{
    int i = blockIdx.x * blockDim.x + threadIdx.x;
    if (i < OCP_ * INC_) {
        int o = i >> 8;                 // /256
        wpad[i] = (o < OC_) ? dn_w[i] : 0.0f;
    }
    if (i < OCP_) bpad[i] = (i < OC_) ? dn_b[i] : 0.0f;
}

// ---------------------------------------------------------------------------
// Prologue: repack conv weights OIHW -> [o][tap][ci] for contiguous K runs
// ---------------------------------------------------------------------------
__global__ void lss_pack_convw(const float* __restrict__ w, float* __restrict__ wp,
                               int Cout, int Cin)
{
    int i = blockIdx.x * blockDim.x + threadIdx.x;
    int n = Cout * Cin * 9;
    if (i >= n) return;
    int o   = i / (Cin * 9);
    int rem = i - o * (Cin * 9);
    int t   = rem / Cin;
    int ci  = rem - t * Cin;
    wp[i] = w[(size_t)(o * Cin + ci) * 9 + t];
}

__global__ void lss_zero_f32(float* __restrict__ p, int n)
{
    int i = blockIdx.x * blockDim.x + threadIdx.x;
    if (i < n) p[i] = 0.0f;
}

// ---------------------------------------------------------------------------
// Kernel 1: depthnet 1x1 conv as GEMM.  D(16x16) += A(16x4) x B(4x16).
// A = wpad[o][k] (112x256 zero-padded), B = x[cam][k][pixel], K=256.
// One wave32 per TWO 16x16 tiles (shared A); all loads & stores unconditional.
// y is written with padded row count OCP_ (rows 105..111 are garbage, unread).
// ---------------------------------------------------------------------------
__global__ void __launch_bounds__(32)
lss_depthnet_gemm(const float* __restrict__ x, const float* __restrict__ wpad,
                  const float* __restrict__ bpad, float* __restrict__ y)
{
    const int lane = threadIdx.x;
    const int m0   = lane & 15;
    const int kq   = lane >> 4;
    const int obase = blockIdx.x * 16;           // 0..96
    const int p0    = blockIdx.y * 32 + m0;      // 175*32 = 5600 exactly
    const int cam   = blockIdx.z;

    const float* xcol = x + (size_t)cam * INC_ * HW_ + p0;
    const float* wrow = wpad + (size_t)(obase + m0) * INC_;

    v8f acc0 = {}, acc1 = {};
    #pragma unroll 4
    for (int k = 0; k < INC_; k += 4) {
        const int ka = k + (kq << 1);
        v2f a, b0, b1;
        a.x  = wrow[ka];
        a.y  = wrow[ka + 1];
        b0.x = xcol[(size_t)ka * HW_];
        b0.y = xcol[(size_t)(ka + 1) * HW_];
        b1.x = xcol[(size_t)ka * HW_ + 16];
        b1.y = xcol[(size_t)(ka + 1) * HW_ + 16];
        acc0 = __builtin_amdgcn_wmma_f32_16x16x4_f32(
                   false, a, false, b0, (short)0, acc0, false, false);
        acc1 = __builtin_amdgcn_wmma_f32_16x16x4_f32(
                   false, a, false, b1, (short)0, acc1, false, false);
    }

    #pragma unroll
    for (int r = 0; r < 8; ++r) {
        const int o = obase + r + (kq << 3);
        const float bias = bpad[o];
        float* yrow = y + ((size_t)cam * OCP_ + o) * HW_ + p0;
        yrow[0]  = acc0[r] + bias;
        yrow[16] = acc1[r] + bias;
    }
}

// ---------------------------------------------------------------------------
// Kernel 2: softmax over depth bins + frustum geometry + splat into PADDED bev.
// One 64-thread block per camera pixel; thread = context channel.
// bev_pad layout: [c*NZ_+z][35*35], interior at +POFF_.
// ---------------------------------------------------------------------------
__global__ void __launch_bounds__(64)
lss_softmax_splat(const float* __restrict__ y, const float* __restrict__ rots,
                  const float* __restrict__ trans, float* __restrict__ bev_pad)
{
    __shared__ float sdep[DD_];
    __shared__ int   svox[DD_];

    const int pid = blockIdx.x;
    const int cam = pid / HW_;
    const int p   = pid - cam * HW_;
    const int h   = p / W_;
    const int w   = p - h * W_;
    const int tid = threadIdx.x;

    const float* yl = y + (size_t)cam * OCP_ * HW_ + p;   // padded row stride

    if (tid < DD_) {
        sdep[tid] = yl[tid * HW_];
        const float dval = 4.0f + (float)tid;
        const float u = (float)w * (1599.0f / 99.0f);   // linspace(0,1599,100)
        const float v = (float)h * (899.0f / 55.0f);    // linspace(0,899,56)
        const float px = u * dval, py = v * dval, pz = dval;
        const float* R = rots + cam * 9;
        const float* T = trans + cam * 3;
        const float gx = R[0] * px + R[1] * py + R[2] * pz + T[0];
        const float gy = R[3] * px + R[4] * py + R[5] * pz + T[1];
        const float gz = R[6] * px + R[7] * py + R[8] * pz + T[2];
        const int cx = (int)((gx + 50.0f) / 3.0f);      // trunc toward zero
        const int cy = (int)((gy + 50.0f) / 3.0f);
        const int cz = (int)((gz + 5.0f)  / 3.0f);
        const bool kept = (cx >= 0) & (cx < NXY_) & (cy >= 0) & (cy < NXY_) &
                          (cz >= 0) & (cz < NZ_);
        // padded voxel offset within a channel pair: z*1225 + (cy+1)*35 + (cx+1)
        svox[tid] = kept ? (cz * PPIX_ + (cy + 1) * PW_ + (cx + 1)) : -1;
    }
    __syncthreads();

    if (tid == 0) {
        float mx = sdep[0];
        for (int i = 1; i < DD_; ++i) mx = fmaxf(mx, sdep[i]);
        float s = 0.0f;
        for (int i = 0; i < DD_; ++i) { float e = expf(sdep[i] - mx); sdep[i] = e; s += e; }
        const float inv = 1.0f / s;
        for (int i = 0; i < DD_; ++i) sdep[i] *= inv;
    }
    __syncthreads();

    const float feat = yl[(DD_ + tid) * HW_];
    float* bc = bev_pad + (size_t)tid * (NZ_ * PPIX_);
    for (int d = 0; d < DD_; ++d) {
        const int vx = svox[d];
        if (vx >= 0) atomicAdd(&bc[vx], sdep[d] * feat);
    }
}

// ---------------------------------------------------------------------------
// Kernel 3: conv3x3 SAME + BN(eval) + ReLU as 9 accumulated GEMMs (one per tap).
// in_pad : [Cin][35][35] zero-ringed.  wp : [Cout][tap][Cin] packed.
// One wave per TWO N-tiles (shared A fragments).  Output padded or dense.
// ---------------------------------------------------------------------------
__global__ void __launch_bounds__(32)
lss_conv3x3_bn_relu(const float* __restrict__ in_pad, const float* __restrict__ wp,
                    const float* __restrict__ bs, const float* __restrict__ bb,
                    const float* __restrict__ bm, const float* __restrict__ bv,
                    float* __restrict__ out, int Cin,
                    int orow_stride, int ocol_stride, int obase_off)
{
    const int lane = threadIdx.x;
    const int m0   = lane & 15;
    const int kq   = lane >> 4;
    const int obase = blockIdx.x * 16;
    const int p0    = blockIdx.y * 32 + m0;
    const int p1    = p0 + 16;
    const bool pok0 = (p0 < NPIX_);
    const bool pok1 = (p1 < NPIX_);
    const int pc0   = pok0 ? p0 : 0;
    const int pc1   = pok1 ? p1 : 0;
    const int hh0   = pc0 / NXY_, ww0 = pc0 - hh0 * NXY_;
    const int hh1   = pc1 / NXY_, ww1 = pc1 - hh1 * NXY_;
    const int in_off0 = hh0 * PW_ + ww0;         // 3x3 window top-left (padded)
    const int in_off1 = hh1 * PW_ + ww1;

    const float* wrow = wp + (size_t)(obase + m0) * 9 * Cin;

    v8f acc0 = {}, acc1 = {};
    #pragma unroll
    for (int t = 0; t < 9; ++t) {
        const int ky = t / 3;                    // compile-time after unroll
        const int kx = t - ky * 3;
        const float* bcol0 = in_pad + in_off0 + ky * PW_ + kx;
        const float* bcol1 = in_pad + in_off1 + ky * PW_ + kx;
        const float* arow  = wrow + t * Cin;
        #pragma unroll 4
        for (int ci = 0; ci < Cin; ci += 4) {
            const int ka = ci + (kq << 1);
            v2f a, b0, b1;
            a.x  = arow[ka];
            a.y  = arow[ka + 1];
            b0.x = bcol0[(size_t)ka * PPIX_];
            b0.y = bcol0[(size_t)(ka + 1) * PPIX_];
            b1.x = bcol1[(size_t)ka * PPIX_];
            b1.y = bcol1[(size_t)(ka + 1) * PPIX_];
            acc0 = __builtin_amdgcn_wmma_f32_16x16x4_f32(
                       false, a, false, b0, (short)0, acc0, false, false);
            acc1 = __builtin_amdgcn_wmma_f32_16x16x4_f32(
                       false, a, false, b1, (short)0, acc1, false, false);
        }
    }

    if (pok0) {
        #pragma unroll
        for (int r = 0; r < 8; ++r) {
            const int o = obase + r + (kq << 3);
            const float inv  = bs[o] * rsqrtf(bv[o] + 1e-5f);
            const float beta = bb[o] - bm[o] * inv;
            const float val  = fmaxf(acc0[r] * inv + beta, 0.0f);
            out[(size_t)o * orow_stride + hh0 * ocol_stride + ww0 + obase_off] = val;
        }
    }
    if (pok1) {
        #pragma unroll
        for (int r = 0; r < 8; ++r) {
            const int o = obase + r + (kq << 3);
            const float inv  = bs[o] * rsqrtf(bv[o] + 1e-5f);
            const float beta = bb[o] - bm[o] * inv;
            const float val  = fmaxf(acc1[r] * inv + beta, 0.0f);
            out[(size_t)o * orow_stride + hh1 * ocol_stride + ww1 + obase_off] = val;
        }
    }
}

// ---------------------------------------------------------------------------
extern "C" void kernel_launch(void* const* d_in, const int* in_sizes, int n_in,
                              void* d_out, int out_size, void* d_ws, size_t ws_size,
                              hipStream_t stream)
{
    (void)in_sizes; (void)n_in; (void)out_size; (void)ws_size;

    const float* x     = (const float*)d_in[0];
    const float* rots  = (const float*)d_in[1];
    const float* trans = (const float*)d_in[2];
    const float* dn_w  = (const float*)d_in[3];
    const float* dn_b  = (const float*)d_in[4];
    const float* bw1   = (const float*)d_in[5];
    const float* bn1_s = (const float*)d_in[6];
    const float* bn1_b = (const float*)d_in[7];
    const float* bn1_m = (const float*)d_in[8];
    const float* bn1_v = (const float*)d_in[9];
    const float* bw2   = (const float*)d_in[10];
    const float* bn2_s = (const float*)d_in[11];
    const float* bn2_b = (const float*)d_in[12];
    const float* bn2_m = (const float*)d_in[13];
    const float* bn2_v = (const float*)d_in[14];
    const float* bw3   = (const float*)d_in[15];
    const float* bn3_s = (const float*)d_in[16];
    const float* bn3_b = (const float*)d_in[17];
    const float* bn3_m = (const float*)d_in[18];
    const float* bn3_v = (const float*)d_in[19];
    const float* bw4   = (const float*)d_in[20];
    const float* bn4_s = (const float*)d_in[21];
    const float* bn4_b = (const float*)d_in[22];
    const float* bn4_m = (const float*)d_in[23];
    const float* bn4_v = (const float*)d_in[24];

    float* out = (float*)d_out;

    // ---- workspace layout (floats) ----
    float* y    = (float*)d_ws;                          // 6*112*5600 = 3,763,200
    float* wpad = y    + (size_t)NCAM * OCP_ * HW_;      // 112*256    =    28,672
    float* bpad = wpad + (size_t)OCP_ * INC_;            // 128 (112 used)
    float* bevp = bpad + 128;                            // 128*1225   =   156,800
    float* h1p  = bevp + (size_t)CZ_ * PPIX_;            // 128*1225
    float* h2p  = h1p  + (size_t)CZ_ * PPIX_;            // 512*1225   =   627,200
    float* h3p  = h2p  + (size_t)512 * PPIX_;            // 512*1225
    float* wp1  = h3p  + (size_t)512 * PPIX_;            // 128*128*9  =   147,456
    float* wp2  = wp1  + (size_t)128 * 128 * 9;          // 512*128*9  =   589,824
    float* wp3  = wp2  + (size_t)512 * 128 * 9;          // 512*512*9  = 2,359,296
    float* wp4  = wp3  + (size_t)512 * 512 * 9;          // 256*512*9  = 1,179,648

    // ---- prologue: pack weights, zero padded activation buffers ----
    {
        int n = OCP_ * INC_;
        lss_pack_dnw<<<(n + 255) / 256, 256, 0, stream>>>(dn_w, dn_b, wpad, bpad);
    }
    {
        int n1 = 128 * 128 * 9, n2 = 512 * 128 * 9, n3 = 512 * 512 * 9, n4 = 256 * 512 * 9;
        lss_pack_convw<<<(n1 + 255) / 256, 256, 0, stream>>>(bw1, wp1, 128, 128);
        lss_pack_convw<<<(n2 + 255) / 256, 256, 0, stream>>>(bw2, wp2, 512, 128);
        lss_pack_convw<<<(n3 + 255) / 256, 256, 0, stream>>>(bw3, wp3, 512, 512);
        lss_pack_convw<<<(n4 + 255) / 256, 256, 0, stream>>>(bw4, wp4, 256, 512);
    }
    {
        // bevp..h3p are contiguous: (128+128+512+512)*1225 floats
        int n = (CZ_ + CZ_ + 512 + 512) * PPIX_;
        lss_zero_f32<<<(n + 255) / 256, 256, 0, stream>>>(bevp, n);
    }

    // ---- 1) depthnet GEMM: 7 x 175 x 6 dual-tile waves ----
    lss_depthnet_gemm<<<dim3(OCP_ / 16, HW_ / 32, NCAM), 32, 0, stream>>>(x, wpad, bpad, y);

    // ---- 2) softmax + geometry + lift/splat into padded BEV ----
    lss_softmax_splat<<<NCAM * HW_, 64, 0, stream>>>(y, rots, trans, bevp);

    // ---- 3) bevencode: 4x (conv3x3 + BN + ReLU), dual-N-tile waves ----
    const int nt2 = (NPIX_ + 31) / 32;  // 35 (covers 70 tiles of 16)
    lss_conv3x3_bn_relu<<<dim3(128 / 16, nt2), 32, 0, stream>>>(
        bevp, wp1, bn1_s, bn1_b, bn1_m, bn1_v, h1p, 128, PPIX_, PW_, POFF_);
    lss_conv3x3_bn_relu<<<dim3(512 / 16, nt2), 32, 0, stream>>>(
        h1p, wp2, bn2_s, bn2_b, bn2_m, bn2_v, h2p, 128, PPIX_, PW_, POFF_);
    lss_conv3x3_bn_relu<<<dim3(512 / 16, nt2), 32, 0, stream>>>(
        h2p, wp3, bn3_s, bn3_b, bn3_m, bn3_v, h3p, 512, PPIX_, PW_, POFF_);
    lss_conv3x3_bn_relu<<<dim3(256 / 16, nt2), 32, 0, stream>>>(
        h3p, wp4, bn4_s, bn4_b, bn4_m, bn4_v, out, 512, NPIX_, NXY_, 0);
}